// RTTModel_30709016167172
// MI455X (gfx1250) — compile-verified
//
#include <hip/hip_runtime.h>
#include <math.h>

#define B_   16
#define N_   1024
#define D_   14
#define H_   2
#define HD_  7
#define SEG_ 8

typedef float v2f __attribute__((ext_vector_type(2)));
typedef float v8f __attribute__((ext_vector_type(8)));

// D = A(16x4) * B(4x16) + C, all f32, wave32.
__device__ __forceinline__ v8f wmma4(v2f a, v2f b, v8f c) {
  return __builtin_amdgcn_wmma_f32_16x16x4_f32(false, a, false, b, (short)0, c,
                                               false, false);
}

// ---------------------------------------------------------------- utilities
__global__ void zero_f(float* p, int n) {
  int i = blockIdx.x * 256 + threadIdx.x;
  if (i < n) p[i] = 0.f;
}

__global__ void build_x(const float* __restrict__ pos, const float* __restrict__ ori,
                        const float* __restrict__ mom, const float* __restrict__ ang,
                        const float* __restrict__ cin, float* __restrict__ x,
                        float* __restrict__ cert, int* __restrict__ p0) {
  int i = blockIdx.x * 256 + threadIdx.x;
  if (i >= B_ * N_) return;
  float o0 = ori[i*4+0], o1 = ori[i*4+1], o2 = ori[i*4+2], o3 = ori[i*4+3];
  float nrm = sqrtf(o0*o0 + o1*o1 + o2*o2 + o3*o3);
  float inv = 1.f / fmaxf(nrm, 1e-12f);
  float* xr = x + (size_t)i * 16;
  xr[0] = pos[i*3+0]; xr[1] = pos[i*3+1]; xr[2] = pos[i*3+2];
  xr[3] = o0*inv; xr[4] = o1*inv; xr[5] = o2*inv; xr[6] = o3*inv;
  xr[7] = mom[i*3+0]; xr[8] = mom[i*3+1]; xr[9] = mom[i*3+2];
  xr[10] = ang[i*3+0]; xr[11] = ang[i*3+1]; xr[12] = ang[i*3+2];
  xr[13] = cin[i];
  xr[14] = 0.f; xr[15] = 0.f;
  cert[i] = 0.5f;
  p0[i] = i & (N_ - 1);
}

// Stage this layer's weights/biases into zero-padded 16x16 / 16 layouts so the
// WMMA kernels can fetch B-fragments with plain unconditional v2f loads.
__global__ void prep_w(const float* __restrict__ Wq, const float* __restrict__ Wk,
                       const float* __restrict__ Wv, const float* __restrict__ Wo,
                       const float* __restrict__ bq, const float* __restrict__ bk,
                       const float* __restrict__ bv, const float* __restrict__ bo,
                       float* __restrict__ Wp, float* __restrict__ bp) {
  int tid = threadIdx.x;                 // 256 threads, 1 block
  const float* Ws[4] = {Wq, Wk, Wv, Wo};
  const float* bs[4] = {bq, bk, bv, bo};
  int d = tid >> 4, e = tid & 15;
#pragma unroll
  for (int m = 0; m < 4; ++m) {
    Wp[m * 256 + tid] = (d < D_ && e < D_) ? Ws[m][d * D_ + e] : 0.f;
    if (tid < 16) bp[m * 16 + tid] = (tid < D_) ? bs[m][tid] : 0.f;
  }
}

// -------------------------------------------------- QKV projection (WMMA f32)
// y = x @ W^T + b ; x and W zero-padded to 16 cols -> branch-free loads.
__global__ __launch_bounds__(128) void proj_qkv(
    const float* __restrict__ x, const float* __restrict__ Wp,
    const float* __restrict__ bp,
    float* __restrict__ Q, float* __restrict__ K, float* __restrict__ V) {
  int wave = blockIdx.x * 4 + (threadIdx.x >> 5);   // 0..1023 tiles of 16 rows
  int lane = threadIdx.x & 31;
  int l15 = lane & 15, hi = lane >> 4;
  int rowbase = wave * 16;

  v2f a[4];
  const float* xr = x + (size_t)(rowbase + l15) * 16 + 2 * hi;
#pragma unroll
  for (int ko = 0; ko < 4; ++ko) a[ko] = *(const v2f*)(xr + ko * 4);

  int d = l15;
  const float* wrow = Wp + d * 16 + 2 * hi;          // padded row of W
#pragma unroll
  for (int m = 0; m < 3; ++m) {
    v8f c = {0.f,0.f,0.f,0.f,0.f,0.f,0.f,0.f};
#pragma unroll
    for (int ko = 0; ko < 4; ++ko) {
      v2f b = *(const v2f*)(wrow + m * 256 + ko * 4);
      c = wmma4(a[ko], b, c);
    }
    float bias = bp[m * 16 + d];                     // padded, unconditional
    if (d < D_) {
      int h = d / HD_, hd = d % HD_;
#pragma unroll
      for (int rr = 0; rr < 8; ++rr) {
        int row = rowbase + rr + 8 * hi;
        int b_ = row >> 10, n = row & (N_ - 1);
        float val = c[rr] + bias;
        size_t base = (size_t)(b_ * H_ + h) * N_ + n;
        if (m == 0)      Q[base * 8  + hd] = val;
        else if (m == 1) K[base * 8  + hd] = val;
        else             V[base * 16 + hd] = val;
      }
    }
  }
}

// ------- fused: scores (WMMA) -> online softmax (both heads) -> attn,
//         head-avg entropy -> cert, and Kmat = (avg+eps)^10
__global__ __launch_bounds__(256) void attn_fused_k(
    const float* __restrict__ Q, const float* __restrict__ K,
    float* __restrict__ attn, float* __restrict__ Kmat,
    float* __restrict__ cert) {
  int wave = blockIdx.x * 8 + (threadIdx.x >> 5);   // 0..1023 = (b, itile)
  int lane = threadIdx.x & 31;
  int l15 = lane & 15, hi = lane >> 4;
  int b = wave >> 6, it = wave & 63;
  const float scale = 0.3779644730092272f;          // 1/sqrt(7)

  v2f a0[2], a1[2];
#pragma unroll
  for (int h = 0; h < 2; ++h) {
    const float* qr = Q + ((size_t)(b * H_ + h) * N_ + it * 16 + l15) * 8 + 2 * hi;
    a0[h] = *(const v2f*)(qr);
    a1[h] = *(const v2f*)(qr + 4);
  }

  float m[2][8], s[2][8];
#pragma unroll
  for (int h = 0; h < 2; ++h)
#pragma unroll
    for (int r = 0; r < 8; ++r) { m[h][r] = -1e30f; s[h][r] = 0.f; }

  // pass 1: online row max/sum per head (lane-local over its columns)
  for (int jt = 0; jt < 64; ++jt) {
#pragma unroll
    for (int h = 0; h < 2; ++h) {
      const float* kr = K + ((size_t)(b * H_ + h) * N_ + jt * 16 + l15) * 8 + 2 * hi;
      v2f b0 = *(const v2f*)(kr);
      v2f b1 = *(const v2f*)(kr + 4);
      v8f c = {0.f,0.f,0.f,0.f,0.f,0.f,0.f,0.f};
      c = wmma4(a0[h], b0, c);
      c = wmma4(a1[h], b1, c);
#pragma unroll
      for (int r = 0; r < 8; ++r) {
        float v = c[r] * scale;
        float mn = fmaxf(m[h][r], v);
        s[h][r] = s[h][r] * __expf(m[h][r] - mn) + __expf(v - mn);
        m[h][r] = mn;
      }
    }
  }
  // combine (m,s) across the 16 lanes holding each row (xor stays in half-wave)
#pragma unroll
  for (int off = 1; off < 16; off <<= 1) {
#pragma unroll
    for (int h = 0; h < 2; ++h)
#pragma unroll
      for (int r = 0; r < 8; ++r) {
        float om = __shfl_xor(m[h][r], off, 32);
        float os = __shfl_xor(s[h][r], off, 32);
        float mn = fmaxf(m[h][r], om);
        s[h][r] = s[h][r] * __expf(m[h][r] - mn) + os * __expf(om - mn);
        m[h][r] = mn;
      }
  }
  float sinv[2][8];
#pragma unroll
  for (int h = 0; h < 2; ++h)
#pragma unroll
    for (int r = 0; r < 8; ++r) sinv[h][r] = 1.f / s[h][r];

  float ent[8] = {0.f,0.f,0.f,0.f,0.f,0.f,0.f,0.f};

  // pass 2: recompute scores, emit normalized attn, avg entropy, Kmat
  for (int jt = 0; jt < 64; ++jt) {
    float p[2][8];
#pragma unroll
    for (int h = 0; h < 2; ++h) {
      const float* kr = K + ((size_t)(b * H_ + h) * N_ + jt * 16 + l15) * 8 + 2 * hi;
      v2f b0 = *(const v2f*)(kr);
      v2f b1 = *(const v2f*)(kr + 4);
      v8f c = {0.f,0.f,0.f,0.f,0.f,0.f,0.f,0.f};
      c = wmma4(a0[h], b0, c);
      c = wmma4(a1[h], b1, c);
#pragma unroll
      for (int r = 0; r < 8; ++r) {
        float v = c[r] * scale;
        float pv = __expf(v - m[h][r]) * sinv[h][r];
        p[h][r] = pv;
        attn[((size_t)(b * H_ + h) * N_ + it * 16 + r + 8 * hi) * N_ +
             jt * 16 + l15] = pv;
      }
    }
#pragma unroll
    for (int r = 0; r < 8; ++r) {
      float av = 0.5f * (p[0][r] + p[1][r]);
      float t = av + 1e-10f;
      ent[r] -= av * __logf(t);
      float t2 = t * t, t4 = t2 * t2, t8 = t4 * t4;
      Kmat[((size_t)b * N_ + it * 16 + r + 8 * hi) * N_ + jt * 16 + l15] = t8 * t2;
    }
  }
#pragma unroll
  for (int off = 1; off < 16; off <<= 1)
#pragma unroll
    for (int r = 0; r < 8; ++r) ent[r] += __shfl_xor(ent[r], off, 32);
  if (l15 == 0) {
#pragma unroll
    for (int r = 0; r < 8; ++r) {
      int row = b * N_ + it * 16 + r + 8 * hi;
      float cu = 1.f / (1.f + __expf(ent[r] - 6.931471805599453f)); // sig(logN-H)
      cert[row] = fmaxf(cert[row], cu);
    }
  }
}

// -------------------------------------------------- Sinkhorn (scaling form)
__global__ void sink_init(float* r, float* c) {
  int i = blockIdx.x * 256 + threadIdx.x;
  if (i < B_ * N_) { r[i] = 1.f; c[i] = 1.f; }
}

__global__ __launch_bounds__(256) void sink_row(const float* __restrict__ Kmat,
                                                const float* __restrict__ cvec,
                                                float* __restrict__ rvec) {
  int wave = blockIdx.x * 8 + (threadIdx.x >> 5);   // 0..16383 rows
  int lane = threadIdx.x & 31;
  int b = wave >> 10;
  const float* kr = Kmat + (size_t)wave * N_;
  const float* cv = cvec + b * N_;
  float acc = 0.f;
  for (int t = 0; t < 32; ++t) { int j = lane + 32 * t; acc += kr[j] * cv[j]; }
#pragma unroll
  for (int off = 16; off > 0; off >>= 1) acc += __shfl_xor(acc, off, 32);
  if (lane == 0) {
    float rv = rvec[wave];
    rvec[wave] = rv / (rv * acc + 1e-10f);
  }
}

// column matvec: partials over 8 row-segments (512 blocks) + tiny finalize
__global__ __launch_bounds__(256) void sink_col_part(const float* __restrict__ Kmat,
                                                     const float* __restrict__ rvec,
                                                     float* __restrict__ part) {
  int bx = blockIdx.x;                 // B*4*SEG = 512
  int b = bx >> 5;
  int chunk = (bx >> 3) & 3;
  int seg = bx & (SEG_ - 1);
  int j = chunk * 256 + threadIdx.x;
  const float* kb = Kmat + (size_t)b * N_ * N_;
  const float* rv = rvec + b * N_;
  float acc = 0.f;
  int i0 = seg * (N_ / SEG_);
  for (int i = i0; i < i0 + N_ / SEG_; ++i) acc += rv[i] * kb[(size_t)i * N_ + j];
  part[(size_t)seg * (B_ * N_) + b * N_ + j] = acc;
}

__global__ void sink_col_fin(const float* __restrict__ part,
                             float* __restrict__ cvec) {
  int i = blockIdx.x * 256 + threadIdx.x;   // 0..16383
  float acc = 0.f;
#pragma unroll
  for (int s = 0; s < SEG_; ++s) acc += part[(size_t)s * (B_ * N_) + i];
  float cv = cvec[i];
  cvec[i] = cv / (cv * acc + 1e-10f);
}

// ---------------- argmax_j (K_ij * c_j) and perm composition (first-max tie)
__global__ __launch_bounds__(256) void perm_k(const float* __restrict__ Kmat,
                                              const float* __restrict__ cvec,
                                              const int* __restrict__ pold,
                                              int* __restrict__ pnew) {
  int wave = blockIdx.x * 8 + (threadIdx.x >> 5);   // row = b*N+i
  int lane = threadIdx.x & 31;
  int b = wave >> 10;
  const float* kr = Kmat + (size_t)wave * N_;
  const float* cv = cvec + b * N_;
  float best = -1e30f; int bidx = 0;
  for (int t = 0; t < 32; ++t) {
    int j = lane + 32 * t;
    float v = kr[j] * cv[j];
    if (v > best) { best = v; bidx = j; }
  }
#pragma unroll
  for (int off = 16; off > 0; off >>= 1) {
    float ov = __shfl_xor(best, off, 32);
    int   oi = __shfl_xor(bidx, off, 32);
    if (ov > best || (ov == best && oi < bidx)) { best = ov; bidx = oi; }
  }
  if (lane == 0) pnew[wave] = pold[b * N_ + bidx];
}

// -------------------------------------------------- out = attn @ V (WMMA f32)
__global__ __launch_bounds__(256) void av_k(const float* __restrict__ attn,
                                            const float* __restrict__ V,
                                            float* __restrict__ AV) {
  int wave = blockIdx.x * 8 + (threadIdx.x >> 5);   // 0..2047 = (bh, itile)
  int lane = threadIdx.x & 31;
  int l15 = lane & 15, hi = lane >> 4;
  int bh = wave >> 6, it = wave & 63;
  const float* arow = attn + ((size_t)bh * N_ + it * 16 + l15) * N_ + 2 * hi;
  const float* vb   = V + (size_t)bh * N_ * 16 + l15;
  v8f c0 = {0.f,0.f,0.f,0.f,0.f,0.f,0.f,0.f};
  v8f c1 = {0.f,0.f,0.f,0.f,0.f,0.f,0.f,0.f};
  for (int kb = 0; kb < N_; kb += 8) {
    v2f a0 = *(const v2f*)(arow + kb);
    v2f a1 = *(const v2f*)(arow + kb + 4);
    int k0 = kb + 2 * hi;
    v2f b0, b1;
    b0.x = vb[(size_t)(k0 + 0) * 16];
    b0.y = vb[(size_t)(k0 + 1) * 16];
    b1.x = vb[(size_t)(k0 + 4) * 16];
    b1.y = vb[(size_t)(k0 + 5) * 16];
    c0 = wmma4(a0, b0, c0);
    c1 = wmma4(a1, b1, c1);
  }
  int b_ = bh >> 1, h = bh & 1;
  int col = l15;
  if (col < HD_) {
#pragma unroll
    for (int rr = 0; rr < 8; ++rr) {
      int n = it * 16 + rr + 8 * hi;
      AV[((size_t)b_ * N_ + n) * 16 + h * HD_ + col] = c0[rr] + c1[rr];
    }
  }
}

// -------------------------------------------------- x' = AV @ Wo^T + bo
__global__ __launch_bounds__(128) void oproj_k(const float* __restrict__ AV,
                                               const float* __restrict__ Wp,
                                               const float* __restrict__ bp,
                                               float* __restrict__ xn) {
  int wave = blockIdx.x * 4 + (threadIdx.x >> 5);
  int lane = threadIdx.x & 31;
  int l15 = lane & 15, hi = lane >> 4;
  int rowbase = wave * 16;
  const float* ar = AV + (size_t)(rowbase + l15) * 16 + 2 * hi;
  int d = l15;
  const float* wrow = Wp + 3 * 256 + d * 16 + 2 * hi;   // padded Wo row
  v8f c = {0.f,0.f,0.f,0.f,0.f,0.f,0.f,0.f};
#pragma unroll
  for (int ko = 0; ko < 4; ++ko) {
    v2f a = *(const v2f*)(ar + ko * 4);
    v2f b = *(const v2f*)(wrow + ko * 4);
    c = wmma4(a, b, c);
  }
  float bias = bp[3 * 16 + d];                          // padded, unconditional
#pragma unroll
  for (int rr = 0; rr < 8; ++rr) {
    int row = rowbase + rr + 8 * hi;
    xn[(size_t)row * 16 + d] = (d < D_) ? (c[rr] + bias) : 0.f;
  }
}

// -------------------------------------------------- outputs: x, cert, perm
__global__ void final_k(const float* __restrict__ x, const float* __restrict__ cert,
                        const int* __restrict__ perm, float* __restrict__ out) {
  int i = blockIdx.x * 256 + threadIdx.x;
  if (i >= B_ * N_) return;
  for (int d = 0; d < D_; ++d) out[(size_t)i * D_ + d] = x[(size_t)i * 16 + d];
  out[B_ * N_ * D_ + i] = cert[i];
  ((int*)out)[B_ * N_ * (D_ + 1) + i] = perm[i];
}

extern "C" void kernel_launch(void* const* d_in, const int* in_sizes, int n_in,
                              void* d_out, int out_size, void* d_ws, size_t ws_size,
                              hipStream_t stream) {
  (void)in_sizes; (void)n_in; (void)out_size; (void)ws_size;
  const float* pos  = (const float*)d_in[0];
  const float* ori  = (const float*)d_in[1];
  const float* mom  = (const float*)d_in[2];
  const float* ang  = (const float*)d_in[3];
  const float* cin  = (const float*)d_in[4];

  float* ws    = (float*)d_ws;
  float* xA    = ws;                                  // 16*1024*16
  float* xB    = xA + 262144;
  float* Qb    = xB + 262144;                         // (B,H,N,8)
  float* Kb    = Qb + 262144;
  float* Vb    = Kb + 262144;                         // (B,H,N,16) padded
  float* AV    = Vb + 524288;                         // (B,N,16) padded
  float* attn  = AV + 262144;                         // (B,H,N,N) 128MB
  float* Kmat  = attn + (size_t)B_ * H_ * N_ * N_;    // (B,N,N)   64MB
  float* certb = Kmat + (size_t)B_ * N_ * N_;
  float* rvec  = certb + B_ * N_;
  float* cvec  = rvec + B_ * N_;
  float* part  = cvec + B_ * N_;                      // SEG * B*N
  float* Wpad  = part + SEG_ * B_ * N_;               // 4 * 16*16
  float* bpad  = Wpad + 4 * 256;                      // 4 * 16
  int*   p0    = (int*)(bpad + 4 * 16);
  int*   p1    = p0 + B_ * N_;

  // Zero the padded Q/K/V/AV region once: pad lanes are never re-written.
  zero_f<<<(1310720 + 255) / 256, 256, 0, stream>>>(Qb, 1310720);
  build_x<<<(B_ * N_ + 255) / 256, 256, 0, stream>>>(pos, ori, mom, ang, cin,
                                                     xA, certb, p0);

  float* xc = xA; float* xn = xB;
  int* pc = p0; int* pn = p1;
  for (int l = 0; l < 4; ++l) {
    const float* Wq = (const float*)d_in[5] + l * D_ * D_;
    const float* Wk = (const float*)d_in[6] + l * D_ * D_;
    const float* Wv = (const float*)d_in[7] + l * D_ * D_;
    const float* Wo = (const float*)d_in[8] + l * D_ * D_;
    const float* bq = (const float*)d_in[9]  + l * D_;
    const float* bk = (const float*)d_in[10] + l * D_;
    const float* bv = (const float*)d_in[11] + l * D_;
    const float* bo = (const float*)d_in[12] + l * D_;

    prep_w<<<1, 256, 0, stream>>>(Wq, Wk, Wv, Wo, bq, bk, bv, bo, Wpad, bpad);
    proj_qkv<<<256, 128, 0, stream>>>(xc, Wpad, bpad, Qb, Kb, Vb);
    attn_fused_k<<<128, 256, 0, stream>>>(Qb, Kb, attn, Kmat, certb);

    sink_init<<<(B_ * N_ + 255) / 256, 256, 0, stream>>>(rvec, cvec);
    for (int it = 0; it < 20; ++it) {
      sink_row<<<2048, 256, 0, stream>>>(Kmat, cvec, rvec);
      sink_col_part<<<B_ * 4 * SEG_, 256, 0, stream>>>(Kmat, rvec, part);
      sink_col_fin<<<64, 256, 0, stream>>>(part, cvec);
    }
    perm_k<<<2048, 256, 0, stream>>>(Kmat, cvec, pc, pn);

    av_k<<<256, 256, 0, stream>>>(attn, Vb, AV);
    oproj_k<<<256, 128, 0, stream>>>(AV, Wpad, bpad, xn);

    float* tx = xc; xc = xn; xn = tx;
    int* tp = pc; pc = pn; pn = tp;
  }
  final_k<<<(B_ * N_ + 255) / 256, 256, 0, stream>>>(xc, certb, pc, (float*)d_out);
}